// LiteratiQuantEmbedding_61838939127938
// MI455X (gfx1250) — compile-verified
//
#include <hip/hip_runtime.h>
#include <hip/hip_bf16.h>

// LiteratiQuantEmbedding: out[token, :] = sign(weight[id,:]) * max(scales[id, d/128], 1e-8)
// Pure gather + elementwise transform. Bandwidth bound: ~128 MiB total -> ~5.5us @ 23.3 TB/s.
// CDNA5 strategy: per-lane async global->LDS staging (ASYNCcnt) with a 4-deep row pipeline,
// non-temporal streamed stores for the output.

#define DIM 1024
#define NGROUP 8            // DIM / 128 scale groups
constexpr int THREADS = 256;        // 8 wave32 per block; each thread owns 4 floats of a row
constexpr int ROWS_PER_BLOCK = 16;  // contiguous token chunk per block
constexpr int PIPE = 4;             // async pipeline depth (4 x 4KB LDS = 16KB / block)

typedef float v4f __attribute__((ext_vector_type(4)));
typedef int   v4i __attribute__((ext_vector_type(4)));

// ---- s_wait_asynccnt <= n ---------------------------------------------------
#if __has_builtin(__builtin_amdgcn_s_wait_asynccnt)
#define WAIT_ASYNC(n) __builtin_amdgcn_s_wait_asynccnt(n)
#else
#define WAIT_ASYNC(n) asm volatile("s_wait_asynccnt " #n ::: "memory")
#endif
// compiler-level reordering fence so the ds_load of async-written LDS stays after the wait
#define OPT_FENCE() asm volatile("" ::: "memory")

// ---- async 16B global -> LDS copy (per lane) --------------------------------
// Builtin signature (from hipcc diagnostic): param 1 = int4 * __device__ (AS1 global),
// so order is (global src, lds dst, imm offset, imm cpol).
__device__ __forceinline__ void async_copy16(const float* gsrc, float* lds_dst) {
#if __has_builtin(__builtin_amdgcn_global_load_async_to_lds_b128)
  __builtin_amdgcn_global_load_async_to_lds_b128(
      (__attribute__((address_space(1))) v4i*)gsrc,
      (__attribute__((address_space(3))) v4i*)lds_dst,
      /*offset=*/0, /*cpol=*/0);
#else
  unsigned loff = (unsigned)(unsigned long long)
      (__attribute__((address_space(3))) float*)lds_dst;
  asm volatile("global_load_async_to_lds_b128 %0, %1, off"
               :: "v"(loff), "v"(gsrc) : "memory");
#endif
}

// sign(0) -> +1 per reference (jnp.where(sign==0, 1, sign)); -0.0 == 0.0 handles both zeros.
__device__ __forceinline__ float qsel(float w, float s) {
  float r = __builtin_copysignf(s, w);
  return (w == 0.0f) ? s : r;
}

__device__ __forceinline__ void consume_row(const float* buf_row, int id,
                                            const float* __restrict__ scales,
                                            float* __restrict__ out_row, int t) {
  const int g = t >> 5;  // 4 floats/thread, 128 floats/group -> group = t/32
  const float s = fmaxf(scales[(size_t)id * NGROUP + g], 1e-8f);
  const v4f wv = *(const v4f*)(buf_row + 4 * t);   // ds_load_b128 (self-staged bytes)
  v4f o;
  o.x = qsel(wv.x, s);
  o.y = qsel(wv.y, s);
  o.z = qsel(wv.z, s);
  o.w = qsel(wv.w, s);
  __builtin_nontemporal_store(o, (v4f*)(out_row + 4 * t));  // streamed: don't pollute L2
}

__global__ __launch_bounds__(THREADS) void literati_gather_quant(
    const int* __restrict__ ids, const float* __restrict__ weight,
    const float* __restrict__ scales, float* __restrict__ out, int nrows) {
  __shared__ float buf[PIPE][DIM];
  const int t = threadIdx.x;
  const int r0 = blockIdx.x * ROWS_PER_BLOCK;

  if (r0 + ROWS_PER_BLOCK <= nrows) {
    // ---- prologue: fill the pipe -------------------------------------------
#pragma unroll
    for (int k = 0; k < PIPE; ++k) {
      const int id = ids[r0 + k];
      async_copy16(weight + (size_t)id * DIM + 4 * t, &buf[k][4 * t]);
    }
    // ---- steady state: PIPE rows in flight ---------------------------------
    // At loop top, rows k..k+PIPE-1 are outstanding (in-order completion):
    // asynccnt <= PIPE-1  ==>  row k has landed in LDS.
#pragma unroll
    for (int k = 0; k < ROWS_PER_BLOCK - PIPE; ++k) {
      WAIT_ASYNC(3);
      OPT_FENCE();
      consume_row(&buf[k % PIPE][0], ids[r0 + k], scales,
                  out + (size_t)(r0 + k) * DIM, t);
      // refill the slot this thread just drained (no cross-thread hazard:
      // each lane reads exactly the 16B it issued)
      const int kn = k + PIPE;
      const int idn = ids[r0 + kn];
      async_copy16(weight + (size_t)idn * DIM + 4 * t, &buf[kn % PIPE][4 * t]);
    }
    // ---- drain: constant wait immediates 3,2,1,0 ---------------------------
    {
      int k = ROWS_PER_BLOCK - 4;
      WAIT_ASYNC(3); OPT_FENCE();
      consume_row(&buf[k % PIPE][0], ids[r0 + k], scales, out + (size_t)(r0 + k) * DIM, t);
      k = ROWS_PER_BLOCK - 3;
      WAIT_ASYNC(2); OPT_FENCE();
      consume_row(&buf[k % PIPE][0], ids[r0 + k], scales, out + (size_t)(r0 + k) * DIM, t);
      k = ROWS_PER_BLOCK - 2;
      WAIT_ASYNC(1); OPT_FENCE();
      consume_row(&buf[k % PIPE][0], ids[r0 + k], scales, out + (size_t)(r0 + k) * DIM, t);
      k = ROWS_PER_BLOCK - 1;
      WAIT_ASYNC(0); OPT_FENCE();
      consume_row(&buf[k % PIPE][0], ids[r0 + k], scales, out + (size_t)(r0 + k) * DIM, t);
    }
  } else {
    // generic tail (unused for the 16384-token config, kept for correctness)
    for (int r = r0; r < nrows; ++r) {
      const int id = ids[r];
      async_copy16(weight + (size_t)id * DIM + 4 * t, &buf[0][4 * t]);
      WAIT_ASYNC(0); OPT_FENCE();
      consume_row(&buf[0][0], id, scales, out + (size_t)r * DIM, t);
    }
  }
}

extern "C" void kernel_launch(void* const* d_in, const int* in_sizes, int n_in,
                              void* d_out, int out_size, void* d_ws, size_t ws_size,
                              hipStream_t stream) {
  const int*   ids    = (const int*)d_in[0];    // input_ids [8,2048] int32
  const float* weight = (const float*)d_in[1];  // [50257, 1024] f32
  const float* scales = (const float*)d_in[2];  // [50257, 8] f32
  float*       out    = (float*)d_out;          // [8, 2048, 1024] f32
  const int nrows = in_sizes[0];                // 16384 tokens
  const int grid = (nrows + ROWS_PER_BLOCK - 1) / ROWS_PER_BLOCK;  // 1024 blocks
  hipLaunchKernelGGL(literati_gather_quant, dim3(grid), dim3(THREADS), 0, stream,
                     ids, weight, scales, out, nrows);
}